// KnlDec_46248207843577
// MI455X (gfx1250) — compile-verified
//
#include <hip/hip_runtime.h>

typedef __attribute__((ext_vector_type(2))) float v2f;
typedef __attribute__((ext_vector_type(8))) float v8f;

#define FC_IN   8192
#define FC_OUT  225
#define FC_PAD  240
#define NK      9
#define KTAPS   25
#define BATCH   32
#define KSPLIT  8
#define KCHUNK  (FC_IN / KSPLIT)   // 1024
#define IMG     256
#define CH      3

// ---------------------------------------------------------------------------
// Kernel 1: FC projection  fc[m, n] = emb[m, :] . W[n, :]
// WMMA f32 16x16x4, one wave per (ntile, mtile, kchunk) block.
// A (16x4 f32): lane l (m = l&15, half = l>>4): VGPR0 = A[m][2*half],
//               VGPR1 = A[m][2*half+1]   (ISA 7.12.2, 32-bit A 16x4)
// B (4x16 f32): symmetric: n = l&15, VGPR0 = B[2*half][n], VGPR1 = B[2*half+1][n]
// C/D (16x16 f32): VGPR r, lane l: row = 8*half + r, col = l&15
// ---------------------------------------------------------------------------
__global__ __launch_bounds__(32)
void fc_wmma_kernel(const float* __restrict__ emb, const float* __restrict__ W,
                    float* __restrict__ fcpart) {
  const int lane  = threadIdx.x;      // 0..31
  const int ntile = blockIdx.x;       // 0..14  (cols 0..239, 225 valid)
  const int mtile = blockIdx.y;       // 0..1   (rows 0..31)
  const int kc    = blockIdx.z;       // 0..7   (K chunk)

  const int m    = lane & 15;
  const int half = lane >> 4;
  const int kb   = half * 2;

  const int arow = mtile * 16 + m;                       // batch row, always valid
  const int ncol = ntile * 16 + m;                       // B column for this lane
  const int ncl  = ncol < FC_OUT ? ncol : (FC_OUT - 1);  // clamp; zero after load
  const bool nvalid = (ncol < FC_OUT);

  const float* aptr = emb + (size_t)arow * FC_IN + kc * KCHUNK + kb;
  const float* bptr = W   + (size_t)ncl  * FC_IN + kc * KCHUNK + kb;

  v8f acc = {0.f, 0.f, 0.f, 0.f, 0.f, 0.f, 0.f, 0.f};
  for (int k = 0; k < KCHUNK; k += 4) {
    v2f a = *(const v2f*)(aptr + k);
    v2f b = *(const v2f*)(bptr + k);
    b.x = nvalid ? b.x : 0.f;   // v_cndmask, keeps EXEC all-1s for WMMA
    b.y = nvalid ? b.y : 0.f;
    acc = __builtin_amdgcn_wmma_f32_16x16x4_f32(
        /*neg_a=*/false, a, /*neg_b=*/false, b,
        /*c_mod=*/(short)0, acc, /*reuse_a=*/false, /*reuse_b=*/false);
  }

  float* outc = fcpart + (size_t)kc * (BATCH * FC_PAD);
  const int col = ntile * 16 + (lane & 15);
#pragma unroll
  for (int r = 0; r < 8; ++r) {
    const int row = mtile * 16 + half * 8 + r;
    outc[row * FC_PAD + col] = acc[r];
  }
}

// ---------------------------------------------------------------------------
// Kernel 2: sum K-partials + bias, relu-shift, normalize each 5x5 kernel.
// 288 threads = one (b, k) pair each. Deterministic fixed-order reduction.
// ---------------------------------------------------------------------------
__global__ __launch_bounds__(288)
void fc_norm_kernel(const float* __restrict__ fcpart, const float* __restrict__ bias,
                    float* __restrict__ kern) {
  const int t  = threadIdx.x;     // 0..287
  const int b  = t / NK;
  const int kk = t % NK;

  float v[KTAPS];
  float s = 0.f;
#pragma unroll
  for (int p = 0; p < KTAPS; ++p) {
    const int col = kk * KTAPS + p;
    float x = bias[col];
    for (int c = 0; c < KSPLIT; ++c)
      x += fcpart[(size_t)c * (BATCH * FC_PAD) + b * FC_PAD + col];
    x = fmaxf(x - 1e-5f, 0.f) + 1e-5f;
    v[p] = x;
    s += x;
  }
#pragma unroll
  for (int p = 0; p < KTAPS; ++p)
    kern[b * (NK * KTAPS) + kk * KTAPS + p] = v[p] / s;
}

// ---------------------------------------------------------------------------
// Kernel 3: per-sample dynamic depthwise 5x5 conv (SAME, cross-correlation).
// Store-bandwidth bound (226 MB out). One thread = one (h,w) pixel of one
// (b,c); patch held in 25 VGPRs, reused across the 9 dynamic kernels.
// Weights are workgroup-uniform -> scalar loads feeding v_fmac.
// ---------------------------------------------------------------------------
__global__ __launch_bounds__(256)
void dyn_conv_kernel(const float* __restrict__ rgb, const float* __restrict__ kern,
                     float* __restrict__ out) {
  const int w = threadIdx.x;     // 0..255
  const int h = blockIdx.x;      // 0..255
  const int c = blockIdx.y;      // 0..2
  const int b = blockIdx.z;      // 0..31

  const float* img = rgb + (size_t)(b * CH + c) * (IMG * IMG);

  float patch[KTAPS];
#pragma unroll
  for (int i = 0; i < 5; ++i) {
    const int hh = h + i - 2;
    const bool hok = (unsigned)hh < (unsigned)IMG;
#pragma unroll
    for (int j = 0; j < 5; ++j) {
      const int ww = w + j - 2;
      const bool ok = hok && ((unsigned)ww < (unsigned)IMG);
      patch[i * 5 + j] = ok ? img[hh * IMG + ww] : 0.f;
    }
  }

  const float* kw = kern + b * (NK * KTAPS);   // uniform across the block
#pragma unroll
  for (int k = 0; k < NK; ++k) {
    float acc = 0.f;
#pragma unroll
    for (int p = 0; p < KTAPS; ++p)
      acc = fmaf(patch[p], kw[k * KTAPS + p], acc);
    out[((size_t)(b * NK + k) * CH + c) * (IMG * IMG) + h * IMG + w] = acc;
  }
}

// ---------------------------------------------------------------------------
extern "C" void kernel_launch(void* const* d_in, const int* in_sizes, int n_in,
                              void* d_out, int out_size, void* d_ws, size_t ws_size,
                              hipStream_t stream) {
  const float* emb  = (const float*)d_in[0];  // [32,128,8,8] -> [32,8192]
  const float* rgb  = (const float*)d_in[1];  // [32,3,256,256]
  const float* W    = (const float*)d_in[2];  // [225,8192]
  const float* bias = (const float*)d_in[3];  // [225]
  float* out = (float*)d_out;                 // [32,9,3,256,256]

  float* fcpart = (float*)d_ws;                              // KSPLIT*32*240 f32
  float* kern   = fcpart + (size_t)KSPLIT * BATCH * FC_PAD;  // 32*225 f32

  fc_wmma_kernel<<<dim3(FC_PAD / 16, BATCH / 16, KSPLIT), 32, 0, stream>>>(emb, W, fcpart);
  fc_norm_kernel<<<1, BATCH * NK, 0, stream>>>(fcpart, bias, kern);
  dyn_conv_kernel<<<dim3(IMG, CH, BATCH), IMG, 0, stream>>>(rgb, kern, out);
}